// GNNEmbedder_68624987456047
// MI455X (gfx1250) — compile-verified
//
#include <hip/hip_runtime.h>
#include <hip/hip_bf16.h>

typedef float v2f __attribute__((ext_vector_type(2)));
typedef float v8f __attribute__((ext_vector_type(8)));

static __device__ __forceinline__ v8f wmma4(v2f a, v2f b, v8f c) {
  // D = A(16x4,f32) * B(4x16,f32) + C(16x16,f32)
  return __builtin_amdgcn_wmma_f32_16x16x4_f32(false, a, false, b, (short)0, c,
                                               false, false);
}

// CDNA5 async memory->LDS DMA (16B per lane), tracked on ASYNCcnt.
static __device__ __forceinline__ void async_copy16_to_lds(unsigned lds_off,
                                                           const void* gptr) {
  asm volatile("global_load_async_to_lds_b128 %0, %1, off"
               :
               : "v"(lds_off), "v"((unsigned long long)(uintptr_t)gptr)
               : "memory");
}
static __device__ __forceinline__ void wait_async0() {
  asm volatile("s_wait_asynccnt 0x0" ::: "memory");
}
// low 32 bits of a flat shared pointer == LDS byte offset (ISA 10.2)
static __device__ __forceinline__ unsigned lds_off_of(const void* p) {
  return (unsigned)(uintptr_t)p;
}

// ---------------------------------------------------------------- utilities
__global__ void k_zero(float* __restrict__ p, int n) {
  int i = blockIdx.x * blockDim.x + threadIdx.x;
  if (i < n) p[i] = 0.0f;
}

// copy 64 columns per row between buffers with different row strides (float4)
__global__ void k_copy_cols(const float* __restrict__ src, int lds_,
                            float* __restrict__ dst, int ldd, int N) {
  int tid = blockIdx.x * blockDim.x + threadIdx.x;
  if (tid >= N * 16) return;
  int row = tid >> 4;
  int c = (tid & 15) << 2;
  *reinterpret_cast<float4*>(&dst[(size_t)row * ldd + c]) =
      *reinterpret_cast<const float4*>(&src[(size_t)row * lds_ + c]);
}

// ------------------------------------------------------- edge scatter (GIN)
// aggr[dst] += h[src]; aggr pre-initialized with h (eps=0 self term).
__global__ void k_edges(const int* __restrict__ ei, int E,
                        const float* __restrict__ h, int ldh,
                        float* __restrict__ aggr) {
  int tid = blockIdx.x * blockDim.x + threadIdx.x;
  int e = tid >> 4;
  if (e >= E) return;
  int f = (tid & 15) << 2;
  int src = ei[e];
  int dst = ei[E + e];
  const float4 v = *reinterpret_cast<const float4*>(&h[(size_t)src * ldh + f]);
  float* d = &aggr[(size_t)dst * 64 + f];
  atomicAdd(d + 0, v.x);
  atomicAdd(d + 1, v.y);
  atomicAdd(d + 2, v.z);
  atomicAdd(d + 3, v.w);
}

// ------------------------------------------- GEMM (A[N,K] @ W[K,64] + bias)
// writes pre-BN Z[N,64] and accumulates per-column sum / sum-of-squares.
__global__ __launch_bounds__(256) void k_gemm_stats(
    const float* __restrict__ A, int lda, int N, int K,
    const float* __restrict__ W, const float* __restrict__ bias,
    float* __restrict__ Z, float* __restrict__ stats) {
  __shared__ float lw[64 * 64];  // one 64-row chunk of W
  const unsigned lwb = lds_off_of(lw);
  const int t = threadIdx.x;
  const int lane = t & 31;
  const int m = lane & 15;   // row (A) / col (B,D) within tile
  const int hi = lane >> 4;  // K-half selector for A/B fragments
  const int row0 = ((blockIdx.x << 3) + (t >> 5)) << 4;  // 8 waves * 16 rows
  const bool active = row0 < N;  // wave-uniform (N % 16 == 0)
  const int row = active ? (row0 + m) : 0;

  v8f acc[4] = {};  // 4 N-tiles of 16 -> 64 output columns

  for (int k0 = 0; k0 < K; k0 += 64) {
    __syncthreads();
    // async DMA one 64x64 chunk of W into LDS: 1024 x 16B, 4 per thread
    const float* Wc = W + (size_t)k0 * 64;
#pragma unroll
    for (int i = 0; i < 4; ++i) {
      int idx = t + i * 256;
      async_copy16_to_lds(lwb + idx * 16, Wc + idx * 4);
    }
    wait_async0();
    __syncthreads();
    if (active) {
      const float* Arow = A + (size_t)row * lda + k0;
#pragma unroll 4
      for (int kk = 0; kk < 64; kk += 4) {
        const int ka = kk + (hi << 1);
        v2f a;
        a.x = Arow[ka];
        a.y = Arow[ka + 1];
#pragma unroll
        for (int nt = 0; nt < 4; ++nt) {
          v2f b;
          b.x = lw[ka * 64 + nt * 16 + m];
          b.y = lw[(ka + 1) * 64 + nt * 16 + m];
          acc[nt] = wmma4(a, b, acc[nt]);
        }
      }
    }
  }
  if (!active) return;
#pragma unroll
  for (int nt = 0; nt < 4; ++nt) {
    const int col = nt * 16 + m;
    const float bv = bias[col];
    float s = 0.0f, sq = 0.0f;
#pragma unroll
    for (int r = 0; r < 8; ++r) {
      float v = acc[nt][r] + bv;
      Z[(size_t)(row0 + r + (hi << 3)) * 64 + col] = v;
      s += v;
      sq += v * v;
    }
    // lanes n and n^16 hold the same column (different row halves):
    // combine cross-lane, then only the low half issues atomics.
    s += __shfl_xor(s, 16, 32);
    sq += __shfl_xor(sq, 16, 32);
    if (hi == 0) {
      atomicAdd(&stats[col], s);
      atomicAdd(&stats[64 + col], sq);
    }
  }
}

// --------------------- BN(stats)+ReLU on A, GEMM with W[64,64], then either
// mode 0: Hout[row*ldo+col] = relu(acc+bias)   (hidden layer)
// mode 1: atomicAdd(pooled[batch[row]*64+col], acc+bias)  (final pool)
__global__ __launch_bounds__(256) void k_bn_gemm(
    const float* __restrict__ Z, int N, const float* __restrict__ stats,
    const float* __restrict__ gamma, const float* __restrict__ beta,
    const float* __restrict__ W, const float* __restrict__ bias,
    float* __restrict__ Hout, int ldo, const int* __restrict__ batch,
    float* __restrict__ pooled, int mode) {
  __shared__ float lw[64 * 64];
  __shared__ float ssc[64], ssh[64];
  const int t = threadIdx.x;
  // kick off async W staging, overlap with BN scale/shift computation
  {
    const unsigned lwb = lds_off_of(lw);
#pragma unroll
    for (int i = 0; i < 4; ++i) {
      int idx = t + i * 256;
      async_copy16_to_lds(lwb + idx * 16, W + idx * 4);
    }
  }
  if (t < 64) {
    const float invN = 1.0f / (float)N;
    const float mu = stats[t] * invN;
    const float var = stats[64 + t] * invN - mu * mu;
    const float sc = rsqrtf(var + 1e-5f) * gamma[t];
    ssc[t] = sc;
    ssh[t] = beta[t] - mu * sc;
  }
  wait_async0();
  __syncthreads();

  const int lane = t & 31;
  const int m = lane & 15;
  const int hi = lane >> 4;
  const int row0 = ((blockIdx.x << 3) + (t >> 5)) << 4;
  if (row0 >= N) return;  // wave-uniform; no barriers after this point
  const int row = row0 + m;

  v8f acc[4] = {};
  const float* Zrow = Z + (size_t)row * 64;
#pragma unroll 4
  for (int kk = 0; kk < 64; kk += 4) {
    const int ka = kk + (hi << 1);
    v2f a;
    a.x = fmaxf(fmaf(Zrow[ka], ssc[ka], ssh[ka]), 0.0f);
    a.y = fmaxf(fmaf(Zrow[ka + 1], ssc[ka + 1], ssh[ka + 1]), 0.0f);
#pragma unroll
    for (int nt = 0; nt < 4; ++nt) {
      v2f b;
      b.x = lw[ka * 64 + nt * 16 + m];
      b.y = lw[(ka + 1) * 64 + nt * 16 + m];
      acc[nt] = wmma4(a, b, acc[nt]);
    }
  }
#pragma unroll
  for (int nt = 0; nt < 4; ++nt) {
    const int col = nt * 16 + m;
    const float bv = bias[col];
#pragma unroll
    for (int r = 0; r < 8; ++r) {
      const int rr = row0 + r + (hi << 3);
      float v = acc[nt][r] + bv;
      if (mode == 0) {
        Hout[(size_t)rr * ldo + col] = fmaxf(v, 0.0f);
      } else {
        atomicAdd(&pooled[(size_t)batch[rr] * 64 + col], v);
      }
    }
  }
}

// ----------------------------------------------------------------- launcher
extern "C" void kernel_launch(void* const* d_in, const int* in_sizes, int n_in,
                              void* d_out, int out_size, void* d_ws,
                              size_t ws_size, hipStream_t stream) {
  const float* x      = (const float*)d_in[0];
  const int*   ei     = (const int*)d_in[1];
  const int*   batch  = (const int*)d_in[2];
  const float* convW1 = (const float*)d_in[3];
  const float* convb1 = (const float*)d_in[4];
  const float* convg  = (const float*)d_in[5];
  const float* convbt = (const float*)d_in[6];
  const float* convW2 = (const float*)d_in[7];
  const float* convb2 = (const float*)d_in[8];
  const float* mlpW1  = (const float*)d_in[9];
  const float* mlpb1  = (const float*)d_in[10];
  const float* mlpg   = (const float*)d_in[11];
  const float* mlpbt  = (const float*)d_in[12];
  const float* mlpW2  = (const float*)d_in[13];
  const float* mlpb2  = (const float*)d_in[14];
  float* out = (float*)d_out;

  const int N = in_sizes[0] / 64;
  const int E = in_sizes[1] / 2;
  const int L = 5, P = 384;  // jump dim = L*H + F

  float* jump  = (float*)d_ws;           // [N, 384]
  float* aggr  = jump + (size_t)N * P;   // [N, 64]
  float* zbuf  = aggr + (size_t)N * 64;  // [N, 64] pre-BN
  float* stats = zbuf + (size_t)N * 64;  // [128] colsum / colsumsq

  const dim3 b256(256);
  const int rowBlocks = (N + 127) / 128;  // 8 waves * 16 rows / block
  const int copyGrid = (N * 16 + 255) / 256;
  const int edgeGrid = (E * 16 + 255) / 256;

  k_copy_cols<<<copyGrid, b256, 0, stream>>>(x, 64, jump, P, N);
  k_zero<<<(out_size + 255) / 256, b256, 0, stream>>>(out, out_size);

  for (int l = 0; l < L; ++l) {
    const float* hl = jump + l * 64;  // layer input, row stride P
    k_zero<<<1, 128, 0, stream>>>(stats, 128);
    k_copy_cols<<<copyGrid, b256, 0, stream>>>(hl, P, aggr, 64, N);
    k_edges<<<edgeGrid, b256, 0, stream>>>(ei, E, hl, P, aggr);
    k_gemm_stats<<<rowBlocks, b256, 0, stream>>>(
        aggr, 64, N, 64, convW1 + l * 4096, convb1 + l * 64, zbuf, stats);
    k_bn_gemm<<<rowBlocks, b256, 0, stream>>>(
        zbuf, N, stats, convg + l * 64, convbt + l * 64, convW2 + l * 4096,
        convb2 + l * 64, jump + (l + 1) * 64, P, nullptr, nullptr, 0);
  }

  // final MLP: jump[N,384] @ mlpW1 -> BN -> ReLU -> @ mlpW2 -> pool by graph
  k_zero<<<1, 128, 0, stream>>>(stats, 128);
  k_gemm_stats<<<rowBlocks, b256, 0, stream>>>(jump, P, N, P, mlpW1, mlpb1,
                                               zbuf, stats);
  k_bn_gemm<<<rowBlocks, b256, 0, stream>>>(zbuf, N, stats, mlpg, mlpbt, mlpW2,
                                            mlpb2, nullptr, 0, batch, out, 1);
}